// PowerSpectrumFeatures_56934086476328
// MI455X (gfx1250) — compile-verified
//
#include <hip/hip_runtime.h>

// CDNA5 / gfx1250. wave32. One wave per atom.
// out[n, :] = concat_l ( (1/sqrt(2l+1)) * (c_l[n]^T @ c_l[n]) .flatten() )
// q = {40,32,24,16}, m = 2l+1 = {1,3,5,7}. Output row stride = 3456 floats.
//
// Memory-bound (~1.1 GB NT stores dominate). Gram tiles via V_WMMA_F32_16X16X4_F32.
// Trick: under the CDNA5 f32 WMMA layouts, the A-fragment (16x4, A = c^T) and
// B-fragment (4x16, B = c) of the same 16-wide tile hold IDENTICAL per-lane
// values: frag.x = c[(2*half + 4*kc) * q + t*16 + (lane&15)]. So one preloaded
// fragment array serves both operands, and all loads are branchless
// (index-clamped + v_cndmask zeroing) to avoid exec-mask save/restore overhead.

typedef __attribute__((ext_vector_type(2))) float v2f;
typedef __attribute__((ext_vector_type(8))) float v8f;

template <int Q, int M>
__device__ __forceinline__ void ps_block(const float* __restrict__ c,  // [M, Q] for this atom
                                         float* __restrict__ out,      // -> Q*Q block
                                         const float scale, const int lane)
{
    constexpr int NT = (Q + 15) / 16;   // 16-wide tiles along q
    constexpr int NK = (M + 3) / 4;     // K-chunks of 4 (zero padded)
    const int half = lane >> 4;         // 0: lanes 0-15, 1: lanes 16-31
    const int lid  = lane & 15;

    // Preload fragments: frag[t][kc].{x,y} = c[(kc*4 + half*2 + {0,1}) * Q + t*16 + lid]
    // (clamped loads are always in-range; OOB lanes zeroed via select, no branches)
    v2f frag[NT][NK];
#pragma unroll
    for (int t = 0; t < NT; ++t) {
        const int idx  = t * 16 + lid;
        const int idxc = (idx < Q) ? idx : (Q - 1);
        const bool iin = (idx < Q);
#pragma unroll
        for (int kc = 0; kc < NK; ++kc) {
            const int k0  = kc * 4 + half * 2;
            const int k0c = (k0     < M) ? k0     : (M - 1);
            const int k1c = (k0 + 1 < M) ? k0 + 1 : (M - 1);
            const float x = c[k0c * Q + idxc];   // unconditional, safe address
            const float y = c[k1c * Q + idxc];
            frag[t][kc].x = (iin && (k0     < M)) ? x : 0.0f;
            frag[t][kc].y = (iin && (k0 + 1 < M)) ? y : 0.0f;
        }
    }

#pragma unroll
    for (int ti = 0; ti < NT; ++ti) {
#pragma unroll
        for (int tj = 0; tj < NT; ++tj) {
            v8f acc = {};
#pragma unroll
            for (int kc = 0; kc < NK; ++kc) {
                // D = A*B + C ; uniform control flow -> EXEC all ones
                acc = __builtin_amdgcn_wmma_f32_16x16x4_f32(
                        /*neg_a=*/false, frag[ti][kc],
                        /*neg_b=*/false, frag[tj][kc],
                        /*c_mod=*/(short)0, acc, /*reuse_a=*/false, /*reuse_b=*/false);
            }
            // D layout: VGPR r -> row = ti*16 + r + 8*half, col = tj*16 + lid
            const int col = tj * 16 + lid;
            if (col < Q) {                       // compile-time true unless ragged tile
#pragma unroll
                for (int r = 0; r < 8; ++r) {
                    const int orow = ti * 16 + r + half * 8;
                    if (orow < Q)
                        __builtin_nontemporal_store(scale * acc[r], &out[orow * Q + col]);
                }
            }
        }
    }
}

__global__ __launch_bounds__(256) void
PowerSpectrumFeatures_kernel(const float* __restrict__ c0,
                             const float* __restrict__ c1,
                             const float* __restrict__ c2,
                             const float* __restrict__ c3,
                             float* __restrict__ out,
                             const int n_atoms)
{
    const int lane = threadIdx.x & 31;
    const int wave = (int)((blockIdx.x * blockDim.x + threadIdx.x) >> 5);
    if (wave >= n_atoms) return;   // wave-uniform exit, EXEC stays all-ones

    float* o = out + (size_t)wave * 3456;

    ps_block<40, 1>(c0 + (size_t)wave * 1 * 40, o +    0, 1.0f,               lane);
    ps_block<32, 3>(c1 + (size_t)wave * 3 * 32, o + 1600, 0.5773502691896258f, lane);
    ps_block<24, 5>(c2 + (size_t)wave * 5 * 24, o + 2624, 0.4472135954999579f, lane);
    ps_block<16, 7>(c3 + (size_t)wave * 7 * 16, o + 3200, 0.3779644730092272f, lane);
}

extern "C" void kernel_launch(void* const* d_in, const int* in_sizes, int n_in,
                              void* d_out, int out_size, void* d_ws, size_t ws_size,
                              hipStream_t stream) {
    const float* c0 = (const float*)d_in[0];   // [N, 1, 40]
    const float* c1 = (const float*)d_in[1];   // [N, 3, 32]
    const float* c2 = (const float*)d_in[2];   // [N, 5, 24]
    const float* c3 = (const float*)d_in[3];   // [N, 7, 16]
    float* out = (float*)d_out;                // [N, 3456]

    const int n_atoms = in_sizes[0] / 40;

    const int threads = 256;                   // 8 waves/block -> 8 atoms/block
    const int waves_per_block = threads / 32;
    const int blocks = (n_atoms + waves_per_block - 1) / waves_per_block;

    PowerSpectrumFeatures_kernel<<<blocks, threads, 0, stream>>>(c0, c1, c2, c3, out, n_atoms);
}